// LengthRegulator_28432683499780
// MI455X (gfx1250) — compile-verified
//
#include <hip/hip_runtime.h>
#include <cstdint>
#include <cstddef>

// LengthRegulator for MI455X (gfx1250, wave32).
// Pure bandwidth problem: ~214 MB traffic, 0 FLOPs -> no WMMA justified.
// CDNA5 paths used: async global->LDS load + s_wait_asynccnt, non-temporal
// 128-bit stores, wave32-exact frame mapping (96 float4 = 32 lanes x 3).

typedef __attribute__((ext_vector_type(4))) float v4f;

constexpr int B  = 16;    // batch
constexpr int T  = 512;   // tokens
constexpr int D  = 384;   // feature dim (= 96 float4 = 32 lanes * 3)
constexpr int ML = 4096;  // max_length

#define GAS __attribute__((address_space(1)))
#define LAS __attribute__((address_space(3)))

// ---------------------------------------------------------------------------
// Kernel 1: per-batch inclusive prefix sum of durations (int32), T threads.
// Durations staged into LDS via CDNA5 async load when the builtin exists.
// ---------------------------------------------------------------------------
__global__ void __launch_bounds__(T)
lr_cumsum_kernel(const int* __restrict__ dur, int* __restrict__ cum) {
  __shared__ int sdur[T];
  __shared__ int sa[T];
  __shared__ int sb[T];

  const int t = threadIdx.x;
  const int b = blockIdx.x;
  const int* gsrc = dur + (size_t)b * T + t;

#if __has_builtin(__builtin_amdgcn_global_load_async_to_lds_b32)
  // Per-lane async copy of the global int at gsrc into LDS at sdur[t]; tracked by ASYNCcnt.
  // Builtin params are typed (int* for b32) per hipcc diagnostic.
  __builtin_amdgcn_global_load_async_to_lds_b32((GAS int*)gsrc,
                                                (LAS int*)&sdur[t],
                                                /*offset=*/0, /*cpol=*/0);
#if __has_builtin(__builtin_amdgcn_s_wait_asynccnt)
  __builtin_amdgcn_s_wait_asynccnt(0);
#else
  asm volatile("s_wait_asynccnt 0" ::: "memory");
#endif
  __syncthreads();
#else
  sdur[t] = *gsrc;
  __syncthreads();
#endif

  sa[t] = sdur[t];
  __syncthreads();

  // Hillis-Steele inclusive scan, ping-pong buffers, 9 rounds for T=512.
  int* src = sa;
  int* dst = sb;
  for (int off = 1; off < T; off <<= 1) {
    int v = src[t];
    if (t >= off) v += src[t - off];
    dst[t] = v;
    __syncthreads();
    int* tmp = src; src = dst; dst = tmp;
  }

  cum[b * T + t] = src[t];
}

// ---------------------------------------------------------------------------
// Kernel 2: one wave per output frame. Wave-uniform upper_bound search over
// cum[b,:], then gather the 1536-byte row with 3 b128 loads per lane and
// write both tuple outputs with non-temporal b128 stores.
// ---------------------------------------------------------------------------
__global__ void __launch_bounds__(256)
lr_gather_kernel(const float* __restrict__ x, const int* __restrict__ cum,
                 float* __restrict__ out) {
  const int lane  = threadIdx.x & 31;
  const int wave  = threadIdx.x >> 5;
  const int frame = blockIdx.x * 8 + wave;   // 8 waves per block
  const int b = frame >> 12;                 // frame / ML   (ML = 4096)
  const int f = frame & (ML - 1);            // frame % ML

  const int* c = cum + b * T;
  const int total = c[T - 1];

  // idx = searchsorted(cum, f, side='right')  (upper_bound)
  int lo = 0, hi = T;
  while (lo < hi) {
    const int mid = (lo + hi) >> 1;
    if (c[mid] <= f) lo = mid + 1; else hi = mid;
  }
  const int idx = (lo < T - 1) ? lo : (T - 1);
  const float m = (f < total) ? 1.0f : 0.0f;

  const v4f* srcRow = (const v4f*)(x + ((size_t)b * T + idx) * D);
  v4f* dst0 = (v4f*)(out + ((size_t)b * ML + f) * D);
  v4f* dst1 = (v4f*)(out + (size_t)B * ML * D + ((size_t)b * ML + f) * D);

#pragma unroll
  for (int k = 0; k < 3; ++k) {
    v4f v = srcRow[lane + 32 * k];   // RT load: rows are reused, keep in L2
    v *= m;                          // out = gathered * mask (matches ref)
    __builtin_nontemporal_store(v, dst0 + lane + 32 * k);  // streamed output
    __builtin_nontemporal_store(v, dst1 + lane + 32 * k);  // mel_pos copy
  }
}

// ---------------------------------------------------------------------------
extern "C" void kernel_launch(void* const* d_in, const int* in_sizes, int n_in,
                              void* d_out, int out_size, void* d_ws,
                              size_t ws_size, hipStream_t stream) {
  (void)in_sizes; (void)n_in; (void)out_size; (void)ws_size;

  const float* x   = (const float*)d_in[0];
  const int*   dur = (const int*)d_in[1];
  // d_in[2] = max_length scalar; compile-time constant ML = 4096 here.

  int* cum = (int*)d_ws;  // B*T ints = 32 KB scratch

  lr_cumsum_kernel<<<B, T, 0, stream>>>(dur, cum);
  lr_gather_kernel<<<(B * ML) / 8, 256, 0, stream>>>(x, cum, (float*)d_out);
}